// Negative_loss_21045339750598
// MI455X (gfx1250) — compile-verified
//
#include <hip/hip_runtime.h>

typedef __attribute__((ext_vector_type(2))) float v2f;
typedef __attribute__((ext_vector_type(4))) float v4f;
typedef __attribute__((ext_vector_type(8))) float v8f;

// Stage 1: each wave streams 8-row groups with two fully-coalesced b128 loads
// (wave-wide 512B each), accumulating per-dim column sums S via
// V_WMMA_F32_16X16X4_F32 (A = ones) and sum-of-squares via VALU FMAs.
// Block writes 33 f32 partials (S[32], sumsq) to workspace.
__global__ __launch_bounds__(256) void neg_loss_partial_kernel(
    const float* __restrict__ X,   // [nRows, 32]
    const float* __restrict__ P,   // [32]
    float* __restrict__ part,      // [nBlocks, 33]
    long nRows) {
  __shared__ float sS[8][64];     // [wave][j*16 + col]
  __shared__ float sQ[8][32];

  const int tid  = threadIdx.x;
  const int L    = tid & 31;      // lane
  const int dcol = (L & 7) << 2;  // this lane's 4 dims: dcol..dcol+3

  // Wave index as a *scalar* so the loop/counter/addresses stay in SGPRs.
  const int w          = __builtin_amdgcn_readfirstlane(tid >> 5);  // 0..7
  const int wavesTotal = (int)gridDim.x * 8;
  const int wvGlobal   = (int)blockIdx.x * 8 + w;

  const v4f pv = *(const v4f*)(P + dcol);

  v8f c0 = {}, c1 = {}, c2 = {}, c3 = {};
  v2f ones; ones.x = 1.0f; ones.y = 1.0f;   // A matrix: all ones (layout-proof)
  float q = 0.0f;

  // Lane-private flat element offset within an 8-row group:
  // lane L covers row 8g + (L>>3), dims dcol..dcol+3 (group0),
  // and row 8g + 4 + (L>>3) (group1, +128 floats).
  const int laneOff = L << 2;

  const int nGroups = (int)(nRows >> 3);    // full 8-row groups

  // ---- guard-free main loop over full groups (scalar counter g) ----
  #pragma unroll 4
  for (int g = wvGlobal; g < nGroups; g += wavesTotal) {
    const float* p0 = X + ((size_t)g << 8) + laneOff;   // g*256 floats
    v4f a0 = *(const v4f*)(p0);          // rows 8g..8g+3   (512B/wave)
    v4f a1 = *(const v4f*)(p0 + 128);    // rows 8g+4..8g+7 (512B/wave)
    v4f d0 = a0 - pv;
    v4f d1 = a1 - pv;

    q += d0.x * d0.x; q += d0.y * d0.y; q += d0.z * d0.z; q += d0.w * d0.w;
    q += d1.x * d1.x; q += d1.y * d1.y; q += d1.z * d1.z; q += d1.w * d1.w;

    v2f b0; b0.x = d0.x; b0.y = d1.x;
    v2f b1; b1.x = d0.y; b1.y = d1.y;
    v2f b2; b2.x = d0.z; b2.y = d1.z;
    v2f b3; b3.x = d0.w; b3.y = d1.w;

    c0 = __builtin_amdgcn_wmma_f32_16x16x4_f32(false, ones, false, b0, (short)0, c0, false, false);
    c1 = __builtin_amdgcn_wmma_f32_16x16x4_f32(false, ones, false, b1, (short)0, c1, false, false);
    c2 = __builtin_amdgcn_wmma_f32_16x16x4_f32(false, ones, false, b2, (short)0, c2, false, false);
    c3 = __builtin_amdgcn_wmma_f32_16x16x4_f32(false, ones, false, b3, (short)0, c3, false, false);
  }

  // ---- remainder rows (< 8): handled by one wave, guarded loads ----
  const long tailBase = (long)nGroups << 3;
  if (wvGlobal == 0 && tailBase < nRows) {   // wave-uniform: EXEC stays all-ones
    const long r0 = tailBase + (L >> 3);
    const long r1 = r0 + 4;
    v4f a0 = (r0 < nRows) ? *(const v4f*)(X + r0 * 32 + dcol) : pv;
    v4f a1 = (r1 < nRows) ? *(const v4f*)(X + r1 * 32 + dcol) : pv;
    v4f d0 = a0 - pv;                        // OOB rows become exactly 0
    v4f d1 = a1 - pv;

    q += d0.x * d0.x; q += d0.y * d0.y; q += d0.z * d0.z; q += d0.w * d0.w;
    q += d1.x * d1.x; q += d1.y * d1.y; q += d1.z * d1.z; q += d1.w * d1.w;

    v2f b0; b0.x = d0.x; b0.y = d1.x;
    v2f b1; b1.x = d0.y; b1.y = d1.y;
    v2f b2; b2.x = d0.z; b2.y = d1.z;
    v2f b3; b3.x = d0.w; b3.y = d1.w;

    c0 = __builtin_amdgcn_wmma_f32_16x16x4_f32(false, ones, false, b0, (short)0, c0, false, false);
    c1 = __builtin_amdgcn_wmma_f32_16x16x4_f32(false, ones, false, b1, (short)0, c1, false, false);
    c2 = __builtin_amdgcn_wmma_f32_16x16x4_f32(false, ones, false, b2, (short)0, c2, false, false);
    c3 = __builtin_amdgcn_wmma_f32_16x16x4_f32(false, ones, false, b3, (short)0, c3, false, false);
  }

  // D rows are identical (A all-ones); VGPR0 lane L holds column (L&15).
  // Column n of c_j is a partial sum of dim 4*(n&7)+j; columns m and m+8
  // together cover all rows of dim 4m+j. Lanes 16-31 are duplicates.
  if (L < 16) {
    sS[w][ 0 + L] = c0[0];
    sS[w][16 + L] = c1[0];
    sS[w][32 + L] = c2[0];
    sS[w][48 + L] = c3[0];
  }
  sQ[w][L] = q;
  __syncthreads();

  if (tid < 32) {                  // thread `tid` produces S[tid]
    const int j = tid & 3;
    const int m = tid >> 2;        // 0..7
    float sAcc = 0.0f, qAcc = 0.0f;
    #pragma unroll
    for (int ww = 0; ww < 8; ++ww) {
      sAcc += sS[ww][j * 16 + m];
      sAcc += sS[ww][j * 16 + m + 8];
      qAcc += sQ[ww][tid];
    }
    part[(long)blockIdx.x * 33 + tid] = sAcc;
    sQ[0][tid] = qAcc;             // own column only: no race with reads above
  }
  __syncthreads();

  if (tid == 0) {
    float qq = 0.0f;
    #pragma unroll
    for (int i = 0; i < 32; ++i) qq += sQ[0][i];
    part[(long)blockIdx.x * 33 + 32] = qq;
  }
}

// Stage 2: one wave sums the per-block partials in fixed order (deterministic)
// and emits the scalar loss = (dot(S,S) - sumsq) / 32.
__global__ __launch_bounds__(32) void neg_loss_finalize_kernel(
    const float* __restrict__ part, int nBlocks, float* __restrict__ out) {
  __shared__ float sh[32];
  const int L = threadIdx.x;
  float s = 0.0f, q = 0.0f;
  for (int b = 0; b < nBlocks; ++b) {
    s += part[(long)b * 33 + L];
    if (L == 0) q += part[(long)b * 33 + 32];
  }
  sh[L] = s * s;
  __syncthreads();
  if (L == 0) {
    float dot = 0.0f;
    #pragma unroll
    for (int i = 0; i < 32; ++i) dot += sh[i];
    out[0] = (dot - q) / 32.0f;
  }
}

extern "C" void kernel_launch(void* const* d_in, const int* in_sizes, int n_in,
                              void* d_out, int out_size, void* d_ws, size_t ws_size,
                              hipStream_t stream) {
  const float* X = (const float*)d_in[0];   // output [N, 32], f32
  const float* P = (const float*)d_in[1];   // pred_avg [32], f32
  float* out = (float*)d_out;               // scalar f32
  float* part = (float*)d_ws;

  const long totalElems = (long)in_sizes[0];
  const long nRows = totalElems / 32;

  // 33 f32 partials per block; clamp block count to available workspace.
  long maxB = (long)(ws_size / (33 * sizeof(float)));
  if (maxB < 1) maxB = 1;
  int nBlocks = (int)(maxB > 1024 ? 1024 : maxB);

  neg_loss_partial_kernel<<<nBlocks, 256, 0, stream>>>(X, P, part, nRows);
  neg_loss_finalize_kernel<<<1, 32, 0, stream>>>(part, nBlocks, out);
}